// GlanceLoss_65712999628875
// MI455X (gfx1250) — compile-verified
//
#include <hip/hip_runtime.h>

#define T_LEN 2048
#define BLOCK 256
#define PER 8
#define ALPHA 0.6f
#define SIGMA 0.2f
#define MIN_MINING_STEP 100

typedef __attribute__((ext_vector_type(2))) float v2f;
typedef __attribute__((ext_vector_type(8))) float v8f;

// ---------- block-wide inclusive scans over an LDS int array of length T_LEN ----------
// 256 threads, each owns PER=8 contiguous elements.

template <typename OpT>
__device__ __forceinline__ void scan_fwd(int* a, int* tot, int ident, OpT op) {
  const int tid = threadIdx.x;
  const int base = tid * PER;
  int run = ident;
#pragma unroll
  for (int k = 0; k < PER; ++k) { run = op(run, a[base + k]); a[base + k] = run; }
  tot[tid] = run;
  __syncthreads();
  for (int off = 1; off < BLOCK; off <<= 1) {
    int v = (tid >= off) ? tot[tid - off] : ident;
    __syncthreads();
    tot[tid] = op(tot[tid], v);
    __syncthreads();
  }
  int ex = (tid > 0) ? tot[tid - 1] : ident;
#pragma unroll
  for (int k = 0; k < PER; ++k) a[base + k] = op(ex, a[base + k]);
  __syncthreads();
}

template <typename OpT>
__device__ __forceinline__ void scan_bwd(int* a, int* tot, int ident, OpT op) {
  const int tid = threadIdx.x;
  const int base = tid * PER;
  int run = ident;
#pragma unroll
  for (int k = PER - 1; k >= 0; --k) { run = op(run, a[base + k]); a[base + k] = run; }
  tot[tid] = run;
  __syncthreads();
  for (int off = 1; off < BLOCK; off <<= 1) {
    int v = (tid + off < BLOCK) ? tot[tid + off] : ident;
    __syncthreads();
    tot[tid] = op(tot[tid], v);
    __syncthreads();
  }
  int ex = (tid < BLOCK - 1) ? tot[tid + 1] : ident;
#pragma unroll
  for (int k = 0; k < PER; ++k) a[base + k] = op(ex, a[base + k]);
  __syncthreads();
}

// ---------- kernel 1: mining + gaussian splat, one block per batch row ----------

__global__ __launch_bounds__(BLOCK)
void glance_mine_splat(const float* __restrict__ scores,
                       const float* __restrict__ labels,
                       const int* __restrict__ step_p,
                       float* __restrict__ target) {
  __shared__ float s_score[T_LEN];   // reused as per-anchor wmin in splat phase
  __shared__ int   s_prev[T_LEN];    // reused as float g-table in splat phase
  __shared__ int   s_nxt[T_LEN];     // reused as anchor index list in splat phase
  __shared__ int   s_scan[T_LEN];    // reused as per-anchor 1/denom in splat phase
  __shared__ unsigned char s_m[T_LEN];
  __shared__ unsigned char s_base[T_LEN];
  __shared__ int   s_tot[BLOCK];
  __shared__ float s_red[BLOCK];
  __shared__ float s_stat[2];
  __shared__ int   s_acount;

  const int tid = threadIdx.x;
  const int row = blockIdx.x;
  const float* sc = scores + row * T_LEN;
  const float* lb = labels + row * T_LEN;
  const int stepv = step_p[0];
  if (tid == 0) s_acount = 0;

  const int cb = tid * PER;
#pragma unroll
  for (int k = 0; k < PER; ++k) {
    int j = cb + k;
    float s = sc[j];
    int mm = (lb[j] > 0.0f) ? 1 : 0;
    s_score[j] = s;
    s_m[j] = (unsigned char)mm;
    s_prev[j] = mm ? j : -1;      // -> cummax = nearest anchor <= j
    s_nxt[j]  = mm ? j : T_LEN;   // -> rev cummin = nearest anchor >= j
  }
  __syncthreads();

  scan_fwd(s_prev, s_tot, -1,    [](int a, int b) { return a > b ? a : b; });
  scan_bwd(s_nxt,  s_tot, T_LEN, [](int a, int b) { return a < b ? a : b; });

  // forward failures: score < alpha * score[prev anchor]
#pragma unroll
  for (int k = 0; k < PER; ++k) {
    int j = cb + k;
    int pv = s_prev[j];
    float sp = s_score[pv < 0 ? 0 : pv];
    int fail = (s_score[j] >= ALPHA * sp) ? 0 : 1;
    s_scan[j] = s_m[j] ? 0 : fail;
  }
  __syncthreads();
  scan_fwd(s_scan, s_tot, 0, [](int a, int b) { return a + b; });

  int minedloc[PER];
#pragma unroll
  for (int k = 0; k < PER; ++k) {
    int j = cb + k;
    int mm = s_m[j];
    int pv = s_prev[j];
    int mf = (pv >= 0 && !mm && (s_scan[j] - s_scan[pv]) == 0) ? 1 : 0;
    minedloc[k] = mm | mf;
  }
  __syncthreads();

  // backward failures (gap threshold reuses alpha*score[prev] per source bug;
  // before-first-anchor segment uses alpha*score[next])
#pragma unroll
  for (int k = 0; k < PER; ++k) {
    int j = cb + k;
    int pv = s_prev[j];
    int nx = s_nxt[j];
    float sp = s_score[pv < 0 ? 0 : pv];
    float sn = s_score[nx > T_LEN - 1 ? T_LEN - 1 : nx];
    float thr = ALPHA * (pv >= 0 ? sp : sn);
    int fail = (s_score[j] >= thr) ? 0 : 1;
    s_scan[j] = s_m[j] ? 0 : fail;
  }
  __syncthreads();
  scan_bwd(s_scan, s_tot, 0, [](int a, int b) { return a + b; });

#pragma unroll
  for (int k = 0; k < PER; ++k) {
    int j = cb + k;
    int mm = s_m[j];
    int nx = s_nxt[j];
    int mb = (nx < T_LEN && !mm && (s_scan[j] - s_scan[nx]) == 0) ? 1 : 0;
    int mined = minedloc[k] | mb;
    s_base[j] = (unsigned char)((stepv < MIN_MINING_STEP) ? mm : mined);
  }
  __syncthreads();

  // ---------------- splat ----------------
  // g(k) = exp(-k^2 * KC), KC = (2/(T-1))^2 / (2*sigma^2); the 1/(sigma*sqrt(2pi))
  // factor cancels in the per-anchor minmax normalization (g(0)=1 is the row max).
  float* s_gtab  = reinterpret_cast<float*>(s_prev);
  int*   s_alist = s_nxt;
  float* s_awmin = s_score;
  float* s_ainv  = reinterpret_cast<float*>(s_scan);
  const float KC = 2.0f / ((float)(T_LEN - 1) * (float)(T_LEN - 1) * SIGMA * SIGMA);

#pragma unroll
  for (int k = 0; k < PER; ++k) {
    int j = k * BLOCK + tid;                     // lane-strided: conflict-free later
    float d = (float)j;
    s_gtab[j] = expf(-(d * d) * KC);
    if (s_base[j]) {
      int pos = atomicAdd(&s_acount, 1);         // order-independent (max later)
      s_alist[pos] = j;
    }
  }
  __syncthreads();

  const int ac = s_acount;
  for (int a = tid; a < ac; a += BLOCK) {
    int p = s_alist[a];
    int kmax = (p > (T_LEN - 1 - p)) ? p : (T_LEN - 1 - p);
    float wmin = s_gtab[kmax];
    float den = 1.0f - wmin;
    s_awmin[a] = wmin;
    s_ainv[a] = (den != 0.0f) ? (1.0f / den) : -1.0f;   // -1 => wn == 1.0 branch
  }
  __syncthreads();

  float tloc[PER];
  float lmin = 3.4e38f, lmax = -3.4e38f;
#pragma unroll
  for (int k = 0; k < PER; ++k) {
    int j = k * BLOCK + tid;
    float t = 0.0f;
    if (ac > 0) {
      t = -3.4e38f;
      for (int a = 0; a < ac; ++a) {
        int p = s_alist[a];
        int d = j - p; if (d < 0) d = -d;
        float g = s_gtab[d];
        float inv = s_ainv[a];
        float wn = (inv > 0.0f) ? (g - s_awmin[a]) * inv : 1.0f;
        t = fmaxf(t, wn);
      }
    }
    tloc[k] = t;
    lmin = fminf(lmin, t);
    lmax = fmaxf(lmax, t);
  }

  // per-row min/max of temp
  s_red[tid] = lmax; __syncthreads();
  for (int off = BLOCK / 2; off > 0; off >>= 1) {
    if (tid < off) s_red[tid] = fmaxf(s_red[tid], s_red[tid + off]);
    __syncthreads();
  }
  if (tid == 0) s_stat[1] = s_red[0];
  __syncthreads();
  s_red[tid] = lmin; __syncthreads();
  for (int off = BLOCK / 2; off > 0; off >>= 1) {
    if (tid < off) s_red[tid] = fminf(s_red[tid], s_red[tid + off]);
    __syncthreads();
  }
  if (tid == 0) s_stat[0] = s_red[0];
  __syncthreads();

  const float tmin = s_stat[0], tmax = s_stat[1];
#pragma unroll
  for (int k = 0; k < PER; ++k) {
    int j = k * BLOCK + tid;
    float t = tloc[k];
    float tn = (tmax != tmin) ? (t - tmin) / (tmax - tmin) : t;
    target[row * T_LEN + j] = (ac > 0) ? tn : 0.0f;
  }
}

// ---------- kernel 2: BCE + mean, WMMA ones-matrix reduction ----------

__global__ __launch_bounds__(BLOCK)
void glance_bce_wmma(const float* __restrict__ scores,
                     const float* __restrict__ target,
                     float* __restrict__ out, int n) {
  __shared__ float red[BLOCK];
  const int tid = threadIdx.x;
  float acc = 0.0f;
  for (int i = tid; i < n; i += BLOCK) {
    float s = scores[i];
    float t = target[i];
    float lp = fmaxf(logf(s), -100.0f);
    float lq = fmaxf(logf(1.0f - s), -100.0f);
    acc -= t * lp + (1.0f - t) * lq;
  }
  red[tid] = acc;
  __syncthreads();

  // Wave 0 only (forced wave-uniform so non-participating waves take the
  // scalar branch and never issue WMMA with EXEC != all-ones).
  int wave = __builtin_amdgcn_readfirstlane((int)(threadIdx.x >> 5));
  if (wave == 0) {
    int lane = tid;            // 0..31
    v2f ones; ones.x = 1.0f; ones.y = 1.0f;   // A = 16x4 all-ones (2 VGPRs/lane)
    v8f c = {};
    for (int it = 0; it < BLOCK / 64; ++it) {
      v2f b;                   // B = 4x16 chunk: 64 partials per WMMA
      b.x = red[it * 64 + lane];
      b.y = red[it * 64 + 32 + lane];
      // D[m,n] += sum_k B[k,n]; with ones-A, sum_n D[0,n] == sum of all 64 B vals
      c = __builtin_amdgcn_wmma_f32_16x16x4_f32(false, ones, false, b,
                                                (short)0, c, false, false);
    }
    float v = c[0];            // row M=0 of D: colsum(n) in lanes 0..15
#pragma unroll
    for (int off = 8; off >= 1; off >>= 1) v += __shfl_xor(v, off, 32);
    if (lane == 0) out[0] = v / (float)n;
  }
}

extern "C" void kernel_launch(void* const* d_in, const int* in_sizes, int n_in,
                              void* d_out, int out_size, void* d_ws, size_t ws_size,
                              hipStream_t stream) {
  (void)n_in; (void)out_size; (void)ws_size;
  const float* scores = (const float*)d_in[0];
  const float* labels = (const float*)d_in[1];
  const int*   step   = (const int*)d_in[2];
  float* target = (float*)d_ws;          // B*T floats of scratch
  float* out    = (float*)d_out;
  const int n = in_sizes[0];             // B*T
  const int rows = n / T_LEN;            // B
  glance_mine_splat<<<rows, BLOCK, 0, stream>>>(scores, labels, step, target);
  glance_bce_wmma<<<1, BLOCK, 0, stream>>>(scores, target, out, n);
}